// CVScaledDotProductAttention_35966056137410
// MI455X (gfx1250) — compile-verified
//
#include <hip/hip_runtime.h>

// Complex-valued scaled-dot-product attention with magnitude min-max norm.
// B=8, H=8, S=1024, D=64, fp32. Uses V_WMMA_F32_16X16X4_F32 (CDNA5 wave32).
//
// Per wave (32 lanes): one 16-row Q strip. Two passes over the 1024-wide K
// axis in 16-column tiles:
//   pass 1: S = (Q/T) K^T (complex, 4 real WMMA chains); track per-row
//           running min/max of |S|; butterfly-reduce across the 16 lanes
//           that share a row in the 16x16 C-layout.
//   pass 2: recompute S (K is L2-resident), normalize elementwise:
//           S' = S * (|S|-mn) / ((mx-mn)*|S|), round-trip S' through padded
//           LDS (C-layout -> A-layout), accumulate complex S' V with WMMA.

typedef __attribute__((ext_vector_type(2))) float v2f;
typedef __attribute__((ext_vector_type(8))) float v8f;

#define NB 8
#define NH 8
#define NS 1024
#define ND 64
#define PLANE (NB * NH * NS * ND)      // 4194304 floats per real/imag plane
#define LDS_STRIDE 20                  // 16 cols + 4 pad: halves hit disjoint banks

__device__ __forceinline__ v8f wmma4(v2f a, v2f b, v8f c) {
  // D = A(16x4,f32) * B(4x16,f32) + C(16x16,f32)
  return __builtin_amdgcn_wmma_f32_16x16x4_f32(
      /*neg_a=*/false, a, /*neg_b=*/false, b,
      /*c_mod=*/(short)0, c, /*reuse_a=*/false, /*reuse_b=*/false);
}

__global__ __launch_bounds__(128) void cv_attn_kernel(
    const float* __restrict__ qr_g, const float* __restrict__ qi_g,
    const float* __restrict__ kr_g, const float* __restrict__ ki_g,
    const float* __restrict__ vr_g, const float* __restrict__ vi_g,
    float* __restrict__ out) {
  __shared__ float lds[4][2][16 * LDS_STRIDE];

  const int lane = threadIdx.x & 31;
  const int wave = threadIdx.x >> 5;
  const int hi   = lane >> 4;   // half-wave (selects K pair / row block)
  const int lo   = lane & 15;   // row (A) or column (B/C/D) within tile
  const int head = blockIdx.y;                    // b*NH + h
  const int qbase = blockIdx.x * 64 + wave * 16;  // this wave's 16 Q rows
  const size_t headOff = (size_t)head * NS * ND;

  // ---- Q fragments (A layout), pre-scaled by 1/TEMPERATURE ----------------
  // A 16x4 f32: lane lo = row M, K index = 4f + 2*hi + j  (j = vector elem)
  v2f aqr[16], aqi[16];
  {
    const float invT = 0.125f;  // 1/8
    const size_t rowOff = headOff + (size_t)(qbase + lo) * ND;
#pragma unroll
    for (int f = 0; f < 16; ++f) {
      const int d0 = 4 * f + 2 * hi;
      aqr[f] = *(const v2f*)(qr_g + rowOff + d0) * invT;
      aqi[f] = *(const v2f*)(qi_g + rowOff + d0) * invT;
    }
  }

  // S-tile for K-tile kt: Sr = qr*kr - qi*ki ; Si = qr*ki + qi*kr
  auto computeS = [&](int kt, v8f& srO, v8f& siO) {
    v8f c1 = {}, c2 = {}, si = {};
    // B 4x16 f32: lane lo = col N (k row kt*16+lo), K index = 4f + 2*hi + j
    const size_t kRowOff = headOff + (size_t)(kt * 16 + lo) * ND;
#pragma unroll
    for (int f = 0; f < 16; ++f) {
      const int d0 = 4 * f + 2 * hi;
      const v2f bkr = *(const v2f*)(kr_g + kRowOff + d0);
      const v2f bki = *(const v2f*)(ki_g + kRowOff + d0);
      c1 = wmma4(aqr[f], bkr, c1);
      c2 = wmma4(aqi[f], bki, c2);
      si = wmma4(aqr[f], bki, si);
      si = wmma4(aqi[f], bkr, si);
    }
    srO = c1 - c2;
    siO = si;
  };

  // ---- pass 1: per-row min/max of |S| over all 1024 columns ---------------
  // C layout: VGPR v in lane (hi,lo) holds element (M = v + 8*hi, N = lo).
  float mn[8], mx[8];
#pragma unroll
  for (int v = 0; v < 8; ++v) { mn[v] = 3.4e38f; mx[v] = -3.4e38f; }

  for (int kt = 0; kt < 64; ++kt) {
    v8f sr, si;
    computeS(kt, sr, si);
#pragma unroll
    for (int v = 0; v < 8; ++v) {
      const float m2 = sr[v] * sr[v] + si[v] * si[v];
      const float mag = __builtin_sqrtf(m2);
      mn[v] = fminf(mn[v], mag);
      mx[v] = fmaxf(mx[v], mag);
    }
  }
  // reduce across the 16 lanes sharing each row (masks < 16 keep hi fixed)
#pragma unroll
  for (int m = 1; m < 16; m <<= 1) {
#pragma unroll
    for (int v = 0; v < 8; ++v) {
      mn[v] = fminf(mn[v], __shfl_xor(mn[v], m, 32));
      mx[v] = fmaxf(mx[v], __shfl_xor(mx[v], m, 32));
    }
  }
  float rsc[8];
#pragma unroll
  for (int v = 0; v < 8; ++v) rsc[v] = 1.0f / (mx[v] - mn[v]);

  // ---- pass 2: recompute S, normalize, accumulate out = S' V (complex) ----
  v8f accR[4] = {{}, {}, {}, {}};  // out real, 4 column tiles of 16 (D=64)
  v8f accI[4] = {{}, {}, {}, {}};  // out imag
  float* ldsR = lds[wave][0];
  float* ldsI = lds[wave][1];

  for (int kt = 0; kt < 64; ++kt) {
    v8f sr, si;
    computeS(kt, sr, si);

    // elementwise: S' = S * (mag - mn) / ((mx - mn) * mag)
#pragma unroll
    for (int v = 0; v < 8; ++v) {
      const float m2  = sr[v] * sr[v] + si[v] * si[v];
      const float inv = __frsqrt_rn(m2);      // 1/mag
      const float mag = m2 * inv;
      const float w   = (mag - mn[v]) * rsc[v] * inv;
      sr[v] *= w;
      si[v] *= w;
    }

    // C-layout -> LDS (padded rows); same-wave DS ops stay in order,
    // compiler inserts s_wait_dscnt for the register dependences.
#pragma unroll
    for (int v = 0; v < 8; ++v) {
      const int m = v + 8 * hi;
      ldsR[m * LDS_STRIDE + lo] = sr[v];
      ldsI[m * LDS_STRIDE + lo] = si[v];
    }

    // A fragments of S' from LDS; B fragments of V from global (L2-hot).
#pragma unroll
    for (int g = 0; g < 4; ++g) {
      const int kk = 4 * g + 2 * hi;  // K index base for this fragment
      const v2f apr  = *(const v2f*)(ldsR + lo * LDS_STRIDE + kk);
      const v2f api  = *(const v2f*)(ldsI + lo * LDS_STRIDE + kk);
      const v2f napi = -api;
#pragma unroll
      for (int c = 0; c < 4; ++c) {
        const size_t vOff = headOff + (size_t)(kt * 16 + kk) * ND + 16 * c + lo;
        v2f bvr, bvi;
        bvr.x = vr_g[vOff];      bvr.y = vr_g[vOff + ND];
        bvi.x = vi_g[vOff];      bvi.y = vi_g[vOff + ND];
        accR[c] = wmma4(apr,  bvr, accR[c]);   // + S'r * Vr
        accR[c] = wmma4(napi, bvi, accR[c]);   // - S'i * Vi
        accI[c] = wmma4(apr,  bvi, accI[c]);   // + S'r * Vi
        accI[c] = wmma4(api,  bvr, accI[c]);   // + S'i * Vr
      }
    }
  }

  // ---- store: out[0] = real plane, out[1] = imag plane --------------------
#pragma unroll
  for (int c = 0; c < 4; ++c) {
#pragma unroll
    for (int v = 0; v < 8; ++v) {
      const int m = v + 8 * hi;
      const size_t idx = headOff + (size_t)(qbase + m) * ND + 16 * c + lo;
      out[idx]         = accR[c][v];
      out[idx + PLANE] = accI[c][v];
    }
  }
}

extern "C" void kernel_launch(void* const* d_in, const int* in_sizes, int n_in,
                              void* d_out, int out_size, void* d_ws, size_t ws_size,
                              hipStream_t stream) {
  (void)in_sizes; (void)n_in; (void)out_size; (void)d_ws; (void)ws_size;
  const float* q_r = (const float*)d_in[0];
  const float* q_i = (const float*)d_in[1];
  const float* k_r = (const float*)d_in[2];
  const float* k_i = (const float*)d_in[3];
  const float* v_r = (const float*)d_in[4];
  const float* v_i = (const float*)d_in[5];
  float* out = (float*)d_out;

  dim3 grid(NS / 64, NB * NH);  // 16 q-blocks x 64 heads
  dim3 block(128);              // 4 waves; 16 q rows per wave
  cv_attn_kernel<<<grid, block, 0, stream>>>(q_r, q_i, k_r, k_i, v_r, v_i, out);
}